// GNN_5463198400657
// MI455X (gfx1250) — compile-verified
//
#include <hip/hip_runtime.h>

#define N_FEAT 32

// ---------------------------------------------------------------------------
// Zero both the intermediate buffer (d_ws) and the output at the start of
// every launch (atomics accumulate; harness does not re-poison between runs).
// ---------------------------------------------------------------------------
__global__ __launch_bounds__(256) void lg_zero2_kernel(float4* __restrict__ a,
                                                       float4* __restrict__ b,
                                                       int n4) {
    int i = blockIdx.x * blockDim.x + threadIdx.x;
    if (i < n4) {
        float4 z = make_float4(0.f, 0.f, 0.f, 0.f);
        a[i] = z;
        b[i] = z;
    }
}

// Non-returning f32 atomic add -> global_atomic_add_f32 (STOREcnt path).
__device__ __forceinline__ void atomic_add_f32(float* p, float v) {
    (void)__hip_atomic_fetch_add(p, v, __ATOMIC_RELAXED, __HIP_MEMORY_SCOPE_AGENT);
}

// Broadcast lane j's value to the whole wave as a *scalar* (SGPR).
// j must be a compile-time constant (fully unrolled loop) -> v_readlane_b32,
// no DS traffic, no dscnt waits.
__device__ __forceinline__ int bcast_i32(int v, int j) {
    return __builtin_amdgcn_readlane(v, j);
}
__device__ __forceinline__ float bcast_f32(float v, int j) {
    return __int_as_float(__builtin_amdgcn_readlane(__float_as_int(v), j));
}

// ---------------------------------------------------------------------------
// One wave32 handles a group of 32 edges.
//   - lane l loads edge (base+l)'s  src / dst / w   (fully coalesced B64/B32)
//   - prefetch the edge stream PF_E edges ahead     (global_prefetch_b8)
//   - unrolled j = 0..31: v_readlane edge j's (s,d,w) into SGPRs,
//     gather 128B of x[s][:] (saddr-form b32), scale by scalar w, and issue
//     32 lane-atomics into out[d][:] (saddr-form global_atomic_add_f32).
// x (12.8 MB) and out (12.8 MB) are fully L2-resident on MI455X (192 MB L2),
// so the random gather + atomic scatter both run at L2 speed.
// ---------------------------------------------------------------------------
#define PF_E 2048  // prefetch distance in edges (~16KB of int64 stream)

__global__ __launch_bounds__(256) void lgconv_scatter_kernel(
    const float* __restrict__ x,
    const long long* __restrict__ src,
    const long long* __restrict__ dst,
    const float* __restrict__ w,
    float* __restrict__ out,
    int n_edges)
{
    const int lane = threadIdx.x & 31;
    const int wave = (blockIdx.x * blockDim.x + threadIdx.x) >> 5;
    const int base = wave * 32;
    if (base >= n_edges) return;

    // Run the sequential edge stream ahead into cache (gfx1250 prefetch path).
    if (base + PF_E + lane < n_edges) {
        __builtin_prefetch(&src[base + PF_E + lane], 0, 1);
        __builtin_prefetch(&dst[base + PF_E + lane], 0, 1);
        __builtin_prefetch(&w[base + PF_E + lane], 0, 1);
    }

    // Coalesced per-lane edge loads: one edge per lane.
    int   sv = 0, dv = 0;
    float wv = 0.f;
    const int my_e = base + lane;
    if (my_e < n_edges) {
        sv = (int)src[my_e];
        dv = (int)dst[my_e];
        wv = w[my_e];
    }

    const float* __restrict__ xl = x + lane;    // lane-fixed voffset
    float* __restrict__       ol = out + lane;

    if (base + 32 <= n_edges) {
        // Full group: straight-line code; readlane gives SGPR (s,d,w) so the
        // compiler can pipeline 32 independent gather->mul->atomic chains.
#pragma unroll
        for (int j = 0; j < 32; ++j) {
            const int   s  = bcast_i32(sv, j);
            const int   d  = bcast_i32(dv, j);
            const float ww = bcast_f32(wv, j);
            const float v  = xl[s * N_FEAT] * ww;
            atomic_add_f32(&ol[d * N_FEAT], v);
        }
    } else {
        // Remainder group (runs at most once per launch): dynamic lane index,
        // use __shfl (ds_bpermute) — negligible.
        const int count = n_edges - base;
        for (int j = 0; j < count; ++j) {
            const int   s  = __shfl(sv, j, 32);
            const int   d  = __shfl(dv, j, 32);
            const float ww = __shfl(wv, j, 32);
            const float v  = xl[s * N_FEAT] * ww;
            atomic_add_f32(&ol[d * N_FEAT], v);
        }
    }
}

extern "C" void kernel_launch(void* const* d_in, const int* in_sizes, int n_in,
                              void* d_out, int out_size, void* d_ws, size_t ws_size,
                              hipStream_t stream) {
    // Inputs (setup_inputs order):
    //   d_in[0]: x           float32  [N, 32]
    //   d_in[1]: edge_index  int64    [2, E]  (row 0 = src, row 1 = dst)
    //   d_in[2]: edge_weight float32  [E]
    const float*     x   = (const float*)d_in[0];
    const long long* ei  = (const long long*)d_in[1];
    const float*     ew  = (const float*)d_in[2];

    const int n_edges = in_sizes[2];                 // 2,500,000
    const int n_feats = in_sizes[0];                 // N * 32 floats

    const long long* src = ei;                       // edge_index[0]
    const long long* dst = ei + n_edges;             // edge_index[1]

    float* out = (float*)d_out;
    float* tmp = (float*)d_ws;                       // intermediate [N, 32]

    // 1) Zero intermediate + output (float4 stores).
    const int n4 = n_feats / 4;
    lg_zero2_kernel<<<(n4 + 255) / 256, 256, 0, stream>>>(
        (float4*)tmp, (float4*)out, n4);

    // 2) Pass 1: tmp[dst] += x[src] * w
    const int n_waves = (n_edges + 31) / 32;
    const int blocks  = (n_waves + 7) / 8;           // 8 wave32 per 256-thread block
    lgconv_scatter_kernel<<<blocks, 256, 0, stream>>>(x, src, dst, ew, tmp, n_edges);

    // 3) Pass 2: out[dst] += tmp[src] * w
    lgconv_scatter_kernel<<<blocks, 256, 0, stream>>>(tmp, src, dst, ew, out, n_edges);
}